// EnergyMovers_82669530513683
// MI455X (gfx1250) — compile-verified
//
#include <hip/hip_runtime.h>
#include <math.h>

typedef __attribute__((ext_vector_type(2))) float v2f;
typedef __attribute__((ext_vector_type(8))) float v8f;

#define BATCH   32
#define NPT     512          // N == M == 512
#define INV_EPS 20.0f        // 1/0.05
#define N_ITERS 50
#define NEGC    (-1.0e9f)
#define MINIT   (-1.0e30f)

struct __align__(16) SmemT {
  float ax[NPT], ay[NPT], ar2[NPT], loga[NPT], logu[NPT], kca[NPT];
  float bx[NPT], by[NPT], br2[NPT], logb[NPT], logv[NPT], kcb[NPT];
  float wred[32];
  float Sa, Sb;
};

__device__ __forceinline__ float wave_sum(float v) {
#pragma unroll
  for (int m = 16; m >= 1; m >>= 1) v += __shfl_xor(v, m, 32);
  return v;
}

// One online-logsumexp sweep:  outLog[n] = outW[n] - lse_m( kc[m]*d(n,m) + inLog[m] )
// out-dim: 16 rows per wave (B operand, constant over tiles); summed dim tiled by 16
// (A operand).  dist via WMMA cross terms: d^2 = r2o + r2s - 2*(o . s).
__device__ __forceinline__ void lse_sweep(
    float* __restrict__ outLog, const float* __restrict__ outW,
    const float* __restrict__ ox, const float* __restrict__ oy,
    const float* __restrict__ or2,
    const float* __restrict__ sx, const float* __restrict__ sy,
    const float* __restrict__ sr2, const float* __restrict__ skc,
    const float* __restrict__ sLog)
{
  const int tid  = threadIdx.x;
  const int lane = tid & 31;
  const int wave = tid >> 5;        // 32 waves -> 32 out-blocks of 16
  const int half = lane >> 4;       // upper lanes carry K=2,3 (zero pad)
  const int l15  = lane & 15;
  const int nIdx = wave * 16 + l15; // this lane's output index

  v2f Bv;                           // B 4x16: K0=x,K1=y rows; upper half zeros
  {
    float bxv = ox[nIdx], byv = oy[nIdx];
    Bv.x = half ? 0.0f : bxv;
    Bv.y = half ? 0.0f : byv;
  }
  const float r2o = or2[nIdx];

  float rm = MINIT, rs = 0.0f;

  for (int t = 0; t < NPT / 16; ++t) {
    const int jb = t * 16;
    float axv = sx[jb + l15], ayv = sy[jb + l15];
    v2f Av;                         // A 16x4: M=summed tile rows
    Av.x = half ? 0.0f : axv;
    Av.y = half ? 0.0f : ayv;
    v8f C = {};
    C = __builtin_amdgcn_wmma_f32_16x16x4_f32(false, Av, false, Bv,
                                              (short)0, C, false, false);

    const int sb = jb + half * 8;   // this lane's 8 summed indices
    const float4 r2a = *(const float4*)(sr2 + sb);
    const float4 r2b = *(const float4*)(sr2 + sb + 4);
    const float4 kc0 = *(const float4*)(skc + sb);
    const float4 kc1 = *(const float4*)(skc + sb + 4);
    const float4 lg0 = *(const float4*)(sLog + sb);
    const float4 lg1 = *(const float4*)(sLog + sb + 4);
    const float r2s[8] = {r2a.x, r2a.y, r2a.z, r2a.w, r2b.x, r2b.y, r2b.z, r2b.w};
    const float kcv[8] = {kc0.x, kc0.y, kc0.z, kc0.w, kc1.x, kc1.y, kc1.z, kc1.w};
    const float lgv[8] = {lg0.x, lg0.y, lg0.z, lg0.w, lg1.x, lg1.y, lg1.z, lg1.w};

    float xv[8];
#pragma unroll
    for (int r = 0; r < 8; ++r) {
      float g  = C[r];              // G[m=sb+r, n=l15]
      float d2 = fmaxf(r2s[r] + r2o - 2.0f * g, 0.0f) + 1e-12f;
      float d  = __builtin_amdgcn_sqrtf(d2);
      xv[r] = fmaf(d, kcv[r], lgv[r]);   // logK + inLog (masked exactly via kc)
    }
    float tm = xv[0];
#pragma unroll
    for (int r = 1; r < 8; ++r) tm = fmaxf(tm, xv[r]);
    float mn = fmaxf(rm, tm);
    float ts = 0.0f;
#pragma unroll
    for (int r = 0; r < 8; ++r) ts += __expf(xv[r] - mn);
    rs = fmaf(rs, __expf(rm - mn), ts);
    rm = mn;
  }
  { // merge lane halves (rows split across lane^16 pairs)
    float om = __shfl_xor(rm, 16, 32);
    float os = __shfl_xor(rs, 16, 32);
    float mn = fmaxf(rm, om);
    rs = rs * __expf(rm - mn) + os * __expf(om - mn);
    rm = mn;
  }
  if (half == 0) {
    float lse = rm + __logf(rs);
    outLog[nIdx] = outW[nIdx] - lse;
  }
}

// Per-thread partial of  sum_{i,j} dist^2 * exp(logu_i + logK_ij + logv_j)
__device__ __forceinline__ float flow_pass(
    const float* __restrict__ lgU,
    const float* __restrict__ ox, const float* __restrict__ oy,
    const float* __restrict__ or2,
    const float* __restrict__ sx, const float* __restrict__ sy,
    const float* __restrict__ sr2, const float* __restrict__ skc,
    const float* __restrict__ lgV)
{
  const int tid  = threadIdx.x;
  const int lane = tid & 31;
  const int wave = tid >> 5;
  const int half = lane >> 4;
  const int l15  = lane & 15;
  const int nIdx = wave * 16 + l15;

  v2f Bv;
  {
    float bxv = ox[nIdx], byv = oy[nIdx];
    Bv.x = half ? 0.0f : bxv;
    Bv.y = half ? 0.0f : byv;
  }
  const float r2o = or2[nIdx];
  const float lu  = lgU[nIdx];

  float acc = 0.0f;
  for (int t = 0; t < NPT / 16; ++t) {
    const int jb = t * 16;
    float axv = sx[jb + l15], ayv = sy[jb + l15];
    v2f Av;
    Av.x = half ? 0.0f : axv;
    Av.y = half ? 0.0f : ayv;
    v8f C = {};
    C = __builtin_amdgcn_wmma_f32_16x16x4_f32(false, Av, false, Bv,
                                              (short)0, C, false, false);

    const int sb = jb + half * 8;
    const float4 r2a = *(const float4*)(sr2 + sb);
    const float4 r2b = *(const float4*)(sr2 + sb + 4);
    const float4 kc0 = *(const float4*)(skc + sb);
    const float4 kc1 = *(const float4*)(skc + sb + 4);
    const float4 lg0 = *(const float4*)(lgV + sb);
    const float4 lg1 = *(const float4*)(lgV + sb + 4);
    const float r2s[8] = {r2a.x, r2a.y, r2a.z, r2a.w, r2b.x, r2b.y, r2b.z, r2b.w};
    const float kcv[8] = {kc0.x, kc0.y, kc0.z, kc0.w, kc1.x, kc1.y, kc1.z, kc1.w};
    const float lgv[8] = {lg0.x, lg0.y, lg0.z, lg0.w, lg1.x, lg1.y, lg1.z, lg1.w};
#pragma unroll
    for (int r = 0; r < 8; ++r) {
      float g  = C[r];
      float d2 = fmaxf(r2s[r] + r2o - 2.0f * g, 0.0f) + 1e-12f;
      float d  = __builtin_amdgcn_sqrtf(d2);
      float fl = __expf(lu + fmaf(d, kcv[r], lgv[r]));  // masked -> exp(-1e9..)=0
      acc = fmaf(d2, fl, acc);
    }
  }
  return acc;
}

__global__ void __launch_bounds__(1024, 1)
emd_sinkhorn_kernel(const float* __restrict__ a_mask, const float* __restrict__ pc_a,
                    const float* __restrict__ b_mask, const float* __restrict__ pc_b,
                    float* __restrict__ out)
{
  __shared__ SmemT s;
  const int b    = blockIdx.x;
  const int tid  = threadIdx.x;
  const int lane = tid & 31;
  const int wave = tid >> 5;

  // ---- setup: threads 0-511 load side A, 512-1023 side B ----
  float pt = 0.0f;
  if (tid < NPT) {
    const int i = tid;
    const float m = a_mask[(size_t)b * NPT + i];
    const float* p = pc_a + ((size_t)b * NPT + i) * 3;
    const float x = p[0], y = p[1], w = p[2];
    s.ax[i] = x; s.ay[i] = y; s.ar2[i] = x * x + y * y;
    pt = m * w;
    s.loga[i] = pt;                               // stash a_pt
    s.kca[i]  = (pt > 0.0f) ? -INV_EPS : 0.0f;    // logK coeff (masked -> 0)
  } else {
    const int i = tid - NPT;
    const float m = b_mask[(size_t)b * NPT + i];
    const float* p = pc_b + ((size_t)b * NPT + i) * 3;
    const float x = p[0], y = p[1], w = p[2];
    s.bx[i] = x; s.by[i] = y; s.br2[i] = x * x + y * y;
    pt = m * w;
    s.logb[i] = pt;                               // stash b_pt
    s.kcb[i]  = (pt > 0.0f) ? -INV_EPS : 0.0f;
    s.logv[i] = 0.0f;                             // Sinkhorn init
  }
  float ws = wave_sum(pt);
  if (lane == 0) s.wred[wave] = ws;
  __syncthreads();
  if (tid == 0) {
    float sa = 0.0f, sb = 0.0f;
    for (int k = 0; k < 16; ++k) sa += s.wred[k];
    for (int k = 16; k < 32; ++k) sb += s.wred[k];
    s.Sa = sa; s.Sb = sb;
  }
  __syncthreads();
  if (tid < NPT) {
    const float apt = s.loga[tid];
    s.loga[tid] = (apt > 0.0f) ? __logf(apt / s.Sa) : NEGC;
  } else {
    const int i = tid - NPT;
    const float bpt = s.logb[i];
    s.logb[i] = (bpt > 0.0f) ? __logf(bpt / s.Sb) : NEGC;
  }
  __syncthreads();

  // ---- 50 Sinkhorn iterations, fully LDS-resident ----
  for (int it = 0; it < N_ITERS; ++it) {
    // logu = loga - lse_j( logK(i,j) + logv_j )
    lse_sweep(s.logu, s.loga, s.ax, s.ay, s.ar2,
              s.bx, s.by, s.br2, s.kcb, s.logv);
    __syncthreads();
    // logv = logb - lse_i( logK(i,j) + logu_i )
    lse_sweep(s.logv, s.logb, s.bx, s.by, s.br2,
              s.ax, s.ay, s.ar2, s.kca, s.logu);
    __syncthreads();
  }

  // ---- ot_term = sum dist^2 * flow ; + huber(Sa - Sb) ----
  float acc  = flow_pass(s.logu, s.ax, s.ay, s.ar2,
                         s.bx, s.by, s.br2, s.kcb, s.logv);
  float wacc = wave_sum(acc);
  if (lane == 0) s.wred[wave] = wacc;
  __syncthreads();
  if (tid == 0) {
    float ot = 0.0f;
    for (int k = 0; k < 32; ++k) ot += s.wred[k];
    const float e  = s.Sa - s.Sb;
    const float ae = fabsf(e);
    const float hub = (ae <= 1.0f) ? 0.5f * e * e : ae - 0.5f;
    out[b] = ot + hub;
  }
}

extern "C" void kernel_launch(void* const* d_in, const int* in_sizes, int n_in,
                              void* d_out, int out_size, void* d_ws, size_t ws_size,
                              hipStream_t stream) {
  (void)in_sizes; (void)n_in; (void)d_ws; (void)ws_size; (void)out_size;
  const float* a_mask = (const float*)d_in[0];
  const float* pc_a   = (const float*)d_in[1];
  const float* b_mask = (const float*)d_in[2];
  const float* pc_b   = (const float*)d_in[3];
  float* out = (float*)d_out;
  emd_sinkhorn_kernel<<<dim3(BATCH), dim3(1024), 0, stream>>>(
      a_mask, pc_a, b_mask, pc_b, out);
}